// GPT_86895778333408
// MI455X (gfx1250) — compile-verified
//
#include <hip/hip_runtime.h>
#include <math.h>

// ---------------------------------------------------------------------------
// B=1, T=2048, C=1024, H=16, d=64, M=8192
// ---------------------------------------------------------------------------
#define T_SEQ   2048
#define C_DIM   1024
#define H_NUM   16
#define D_HEAD  64
#define M_STORE 8192
#define QKV_LD  3072
#define SCALE   0.125f
#define MASK_TH (1.0f / 8192.0f)

typedef __attribute__((ext_vector_type(2))) float v2f;
typedef __attribute__((ext_vector_type(4))) float v4f;
typedef __attribute__((ext_vector_type(8))) float v8f;

static __device__ __forceinline__ int laneid() { return threadIdx.x & 31; }

static __device__ __forceinline__ v8f wmma_f32(v2f a, v2f b, v8f c) {
  return __builtin_amdgcn_wmma_f32_16x16x4_f32(false, a, false, b,
                                               (short)0, c, false, false);
}

// 8 contiguous floats per lane = the lane's share of a 16-wide K block.
// K-permutation: WMMA step s (0..3) uses real k = k0 + 8*half + 2s (+1).
// Both operands of every product use this scheme, so products pair exactly.
struct F8 { v4f lo, hi; };

static __device__ __forceinline__ void load_f8(F8& f, const float* __restrict__ p,
                                               int off) {
  f.lo = *(const v4f*)(p + off);
  f.hi = *(const v4f*)(p + off + 4);
}
static __device__ __forceinline__ v2f f8_frag(const F8& f, int s) {
  v2f r;
  if      (s == 0) { r.x = f.lo.x; r.y = f.lo.y; }
  else if (s == 1) { r.x = f.lo.z; r.y = f.lo.w; }
  else if (s == 2) { r.x = f.hi.x; r.y = f.hi.y; }
  else             { r.x = f.hi.z; r.y = f.hi.w; }
  return r;
}

// C/D 16x16 layout: VGPR i -> row (i + 8*half), col = lane&15
static __device__ __forceinline__ void store_tile(float* __restrict__ o, int ld, v8f c) {
  int l = laneid(); int r = l & 15; int h = l >> 4;
#pragma unroll
  for (int i = 0; i < 8; ++i) o[(i + 8 * h) * ld + r] = c[i];
}

static __device__ __forceinline__ float rowred_max(float v) {
#pragma unroll
  for (int m = 1; m <= 8; m <<= 1) v = fmaxf(v, __shfl_xor(v, m, 32));
  return v;
}
static __device__ __forceinline__ float rowred_sum(float v) {
#pragma unroll
  for (int m = 1; m <= 8; m <<= 1) v += __shfl_xor(v, m, 32);
  return v;
}
static __device__ __forceinline__ float wave_sum(float v) {
#pragma unroll
  for (int m = 1; m <= 16; m <<= 1) v += __shfl_xor(v, m, 32);
  return v;
}

// ---------------------------------------------------------------------------
// K0: |key_store[h][m]|^2
// ---------------------------------------------------------------------------
__global__ __launch_bounds__(256) void ksq_kernel(const float* __restrict__ ks,
                                                  float* __restrict__ ksq, int n) {
  int i = blockIdx.x * 256 + threadIdx.x;
  if (i >= n) return;
  const float* p = ks + (size_t)i * D_HEAD;
  float s = 0.f;
#pragma unroll
  for (int j = 0; j < D_HEAD; j += 4) {
    v4f v = *(const v4f*)(p + j);
    s += v.x * v.x + v.y * v.y + v.z * v.z + v.w * v.w;
  }
  ksq[i] = s;
}

// ---------------------------------------------------------------------------
// K1/K6: O[MxN] = A[MxK] * Bt[NxK]^T.  One wave -> 32x64 tile.  Per-lane base
// pointers fixed before the loop; all in-loop loads use immediate offsets and
// six pointer bumps per 64 WMMAs.  Requires K % 32 == 0.
// ---------------------------------------------------------------------------
__global__ __launch_bounds__(256) void gemm_xwT(const float* __restrict__ A,
                                                const float* __restrict__ Bt,
                                                float* __restrict__ O,
                                                int M, int N, int K,
                                                int lda, int ldb, int ldo) {
  int wave = threadIdx.x >> 5;
  int w = blockIdx.x * 8 + wave;
  int ntN = N >> 6;
  int mt = w / ntN, nt = w % ntN;
  if (mt >= (M >> 5)) return;                 // wave-uniform
  int row0 = mt << 5, col0 = nt << 6;
  int l = laneid(); int r = l & 15; int h = l >> 4;

  const float* pA0 = A + (row0 + r) * lda + 8 * h;
  const float* pA1 = pA0 + 16 * lda;
  const float* pB0 = Bt + (col0 + r) * ldb + 8 * h;
  const float* pB1 = pB0 + 16 * ldb;
  const float* pB2 = pB0 + 32 * ldb;
  const float* pB3 = pB0 + 48 * ldb;

  const v8f vz = {0,0,0,0,0,0,0,0};
  v8f acc[2][4];
#pragma unroll
  for (int mi = 0; mi < 2; ++mi)
#pragma unroll
    for (int j = 0; j < 4; ++j) acc[mi][j] = vz;

  F8 aP[2], bP[4], aQ[2], bQ[4];
  auto LOAD = [&](F8* aa, F8* bb, int ko) {   // ko: compile-time 0 or 16
    load_f8(aa[0], pA0, ko);
    load_f8(aa[1], pA1, ko);
    load_f8(bb[0], pB0, ko);
    load_f8(bb[1], pB1, ko);
    load_f8(bb[2], pB2, ko);
    load_f8(bb[3], pB3, ko);
  };
  auto ADV = [&]() { pA0 += 32; pA1 += 32; pB0 += 32; pB1 += 32; pB2 += 32; pB3 += 32; };
  auto COMP = [&](F8* aa, F8* bb) {
#pragma unroll
    for (int s = 0; s < 4; ++s) {
      v2f af0 = f8_frag(aa[0], s), af1 = f8_frag(aa[1], s);
#pragma unroll
      for (int j = 0; j < 4; ++j) {
        v2f bf = f8_frag(bb[j], s);
        acc[0][j] = wmma_f32(af0, bf, acc[0][j]);
        acc[1][j] = wmma_f32(af1, bf, acc[1][j]);
      }
    }
  };

  LOAD(aP, bP, 0);
  for (int kb = 0; kb < K; kb += 32) {
    LOAD(aQ, bQ, 16);
    COMP(aP, bP);
    ADV();
    if (kb + 32 < K) LOAD(aP, bP, 0);
    COMP(aQ, bQ);
  }
#pragma unroll
  for (int mi = 0; mi < 2; ++mi)
#pragma unroll
    for (int j = 0; j < 4; ++j)
      store_tile(O + (size_t)(row0 + 16 * mi) * ldo + col0 + 16 * j, ldo, acc[mi][j]);
}

// ---------------------------------------------------------------------------
// K2: mask[h][j] = softmax_j(q[T-1].k[j]/8) >= 1/8192   (one block per head)
// ---------------------------------------------------------------------------
__global__ __launch_bounds__(256) void last_att_mask(const float* __restrict__ qkv,
                                                     int* __restrict__ mask) {
  __shared__ float qv[D_HEAD];
  __shared__ float red[256];
  int hh = blockIdx.x, tid = threadIdx.x;
  if (tid < D_HEAD) qv[tid] = qkv[(T_SEQ - 1) * QKV_LD + hh * D_HEAD + tid];
  __syncthreads();
  float s[8]; float mx = -3.4e38f;
#pragma unroll
  for (int jj = 0; jj < 8; ++jj) {
    int j = tid + jj * 256;
    const float* kp = qkv + j * QKV_LD + C_DIM + hh * D_HEAD;
    float dt = 0.f;
#pragma unroll
    for (int c = 0; c < D_HEAD; c += 4) {
      v4f v = *(const v4f*)(kp + c);
      dt += qv[c] * v.x + qv[c + 1] * v.y + qv[c + 2] * v.z + qv[c + 3] * v.w;
    }
    s[jj] = dt * SCALE;
    mx = fmaxf(mx, s[jj]);
  }
  red[tid] = mx; __syncthreads();
  for (int st = 128; st > 0; st >>= 1) {
    if (tid < st) red[tid] = fmaxf(red[tid], red[tid + st]);
    __syncthreads();
  }
  float gmx = red[0]; __syncthreads();
  float ls = 0.f;
#pragma unroll
  for (int jj = 0; jj < 8; ++jj) ls += __expf(s[jj] - gmx);
  red[tid] = ls; __syncthreads();
  for (int st = 128; st > 0; st >>= 1) {
    if (tid < st) red[tid] += red[tid + st];
    __syncthreads();
  }
  float inv = 1.0f / red[0];
#pragma unroll
  for (int jj = 0; jj < 8; ++jj) {
    int j = tid + jj * 256;
    mask[hh * T_SEQ + j] = (__expf(s[jj] - gmx) * inv >= MASK_TH) ? 1 : 0;
  }
}

// ---------------------------------------------------------------------------
// K3: per (head, 16-token tile) stream 8192 store rows through WMMA, rank by
// |ks|^2 - 2 k.ks, lane-private top-4, partner-lane merge.  Double-buffered,
// pointer-bump addressing, L2 prefetch four tiles ahead of the stream.
// ---------------------------------------------------------------------------
__global__ __launch_bounds__(256) void knn_topk(const float* __restrict__ qkv,
                                                const float* __restrict__ kstore,
                                                const float* __restrict__ ksq,
                                                int* __restrict__ idxout) {
  int wave = threadIdx.x >> 5;
  int w = blockIdx.x * 8 + wave;          // 0..2047
  int hh = w >> 7;
  int nt = w & 127;
  int n0 = nt << 4;
  int l = laneid(); int r = l & 15; int hf = l >> 4;
  const float* pA = kstore + (size_t)hh * M_STORE * D_HEAD + r * D_HEAD + 8 * hf;
  const float* pq = ksq + hh * M_STORE + 8 * hf;
  const float* Bt0 = qkv + C_DIM + hh * D_HEAD;          // k-matrix, ld=3072

  F8 bq[4];                                // token fragments: loop invariant
  {
    int boff = (n0 + r) * QKV_LD + 8 * hf;
#pragma unroll
    for (int kk = 0; kk < 4; ++kk) load_f8(bq[kk], Bt0, boff + 16 * kk);
  }

  float bv0 = 3.4e38f, bv1 = 3.4e38f, bv2 = 3.4e38f, bv3 = 3.4e38f;
  int   bi0 = 0, bi1 = 0, bi2 = 0, bi3 = 0;

  F8 aP[4], aQ[4]; v4f kqP0, kqP1, kqQ0, kqQ1;
  auto LOADA = [&](F8* aa, v4f& q0, v4f& q1, int t) {   // t: compile-time 0/1
#pragma unroll
    for (int kk = 0; kk < 4; ++kk) load_f8(aa[kk], pA, t * 16 * D_HEAD + 16 * kk);
    q0 = *(const v4f*)(pq + t * 16);
    q1 = *(const v4f*)(pq + t * 16 + 4);
  };
  auto COMPT = [&](F8* aa, v4f q0, v4f q1, int m0) {
    v8f c = {0,0,0,0,0,0,0,0};
#pragma unroll
    for (int kk = 0; kk < 4; ++kk)
#pragma unroll
      for (int s = 0; s < 4; ++s)
        c = wmma_f32(f8_frag(aa[kk], s), f8_frag(bq[kk], s), c);
#pragma unroll
    for (int i = 0; i < 8; ++i) {
      float kqv = (i < 4) ? q0[i] : q1[i - 4];
      float dv = kqv - 2.0f * c[i];
      int m = m0 + i + 8 * hf;
      if (dv < bv3) {
        if (dv < bv0)      { bv3=bv2;bi3=bi2; bv2=bv1;bi2=bi1; bv1=bv0;bi1=bi0; bv0=dv;bi0=m; }
        else if (dv < bv1) { bv3=bv2;bi3=bi2; bv2=bv1;bi2=bi1; bv1=dv;bi1=m; }
        else if (dv < bv2) { bv3=bv2;bi3=bi2; bv2=dv;bi2=m; }
        else               { bv3=dv;bi3=m; }
      }
    }
  };

  LOADA(aP, kqP0, kqP1, 0);
  for (int mt = 0; mt < (M_STORE >> 4); mt += 2) {
    LOADA(aQ, kqQ0, kqQ1, 1);
    __builtin_prefetch(pA + 4 * 16 * D_HEAD, 0, 1);     // stream hint (+4 tiles)
    COMPT(aP, kqP0, kqP1, mt << 4);
    pA += 2 * 16 * D_HEAD; pq += 32;
    if (mt + 2 < (M_STORE >> 4)) LOADA(aP, kqP0, kqP1, 0);
    COMPT(aQ, kqQ0, kqQ1, (mt + 1) << 4);
  }

  float av[4] = {bv0, bv1, bv2, bv3};
  int   ai[4] = {bi0, bi1, bi2, bi3};
  float ov[4]; int oi[4];
#pragma unroll
  for (int t = 0; t < 4; ++t) { ov[t] = __shfl_xor(av[t], 16, 32); oi[t] = __shfl_xor(ai[t], 16, 32); }
  float mv[4]; int mi[4]; int ia = 0, ib = 0;
#pragma unroll
  for (int t = 0; t < 4; ++t) {
    if (av[ia] <= ov[ib]) { mv[t] = av[ia]; mi[t] = ai[ia]; ++ia; }
    else                  { mv[t] = ov[ib]; mi[t] = oi[ib]; ++ib; }
  }
  (void)mv;
  if (hf == 0) {
    int* o = idxout + (hh * T_SEQ + n0 + r) * 4;
    o[0] = mi[0]; o[1] = mi[1]; o[2] = mi[2]; o[3] = mi[3];
  }
}

// ---------------------------------------------------------------------------
// K4: gather 4 neighbours, 5-way softmax vs q, 0.5/0.5 blend, mask select.
// ---------------------------------------------------------------------------
__global__ __launch_bounds__(256) void knn_vnew(const float* __restrict__ qkv,
                                                const float* __restrict__ kstore,
                                                const float* __restrict__ vstore,
                                                const int* __restrict__ idx,
                                                const int* __restrict__ mask,
                                                float* __restrict__ vnew) {
  int wave = threadIdx.x >> 5;
  int w = blockIdx.x * 8 + wave;          // 0..32767
  int hh = w >> 11; int n = w & (T_SEQ - 1);
  int lane = laneid();
  const float* qp = qkv + n * QKV_LD + hh * D_HEAD + lane * 2;
  v2f q  = *(const v2f*)qp;
  v2f kh = *(const v2f*)(qp + C_DIM);
  v2f vh = *(const v2f*)(qp + 2 * C_DIM);
  float att[5], vx[5], vy[5];
  vx[0] = vh.x; vy[0] = vh.y;
  att[0] = wave_sum(kh.x * q.x + kh.y * q.y) * SCALE;
  const int* ix = idx + (hh * T_SEQ + n) * 4;
#pragma unroll
  for (int j = 0; j < 4; ++j) {
    int m = ix[j];
    v2f kf = *(const v2f*)(kstore + ((size_t)hh * M_STORE + m) * D_HEAD + lane * 2);
    v2f vf = *(const v2f*)(vstore + ((size_t)hh * M_STORE + m) * D_HEAD + lane * 2);
    vx[j + 1] = vf.x; vy[j + 1] = vf.y;
    att[j + 1] = wave_sum(kf.x * q.x + kf.y * q.y) * SCALE;
  }
  float amx = att[0];
#pragma unroll
  for (int j = 1; j < 5; ++j) amx = fmaxf(amx, att[j]);
  float e[5], ssum = 0.f;
#pragma unroll
  for (int j = 0; j < 5; ++j) { e[j] = __expf(att[j] - amx); ssum += e[j]; }
  float inv = 1.0f / ssum;
  float ax = 0.f, ay = 0.f;
#pragma unroll
  for (int j = 0; j < 5; ++j) { ax += e[j] * vx[j]; ay += e[j] * vy[j]; }
  ax *= inv; ay *= inv;
  v2f o;
  if (mask[hh * T_SEQ + n]) { o.x = 0.5f * ax + 0.5f * vh.x; o.y = 0.5f * ay + 0.5f * vh.y; }
  else                      { o.x = vh.x;                    o.y = vh.y; }
  *(v2f*)(vnew + ((size_t)hh * T_SEQ + n) * D_HEAD + lane * 2) = o;
}

// ---------------------------------------------------------------------------
// K5: flash attention y = softmax_causal(q k^T/8) @ v_new.  Per-wave 16-query
// tile; K-tile loads for jt+1 issued before the P@V WMMAs; all in-loop V/K
// accesses via per-lane pointers + immediate offsets.
// ---------------------------------------------------------------------------
__global__ __launch_bounds__(256) void flash_attn(const float* __restrict__ qkv,
                                                  const float* __restrict__ vnew,
                                                  float* __restrict__ y) {
  __shared__ float pld[8][16][18];          // 18-float row stride: v2f-aligned
  int wave = threadIdx.x >> 5;
  int w = blockIdx.x * 8 + wave;            // 0..2047
  int hh = w >> 7, qt = w & 127;
  int q0 = qt << 4;
  int l = laneid(); int r = l & 15; int hf = l >> 4;
  float (*P)[18] = pld[wave];

  const v8f vz = {0,0,0,0,0,0,0,0};
  v8f acc[4];
  float mrow[8], lrow[8];
#pragma unroll
  for (int t = 0; t < 4; ++t) acc[t] = vz;
#pragma unroll
  for (int i = 0; i < 8; ++i) { mrow[i] = -3.4e38f; lrow[i] = 0.f; }

  F8 qfb[4], kfb[4];
  {
    const float* pQ = qkv + hh * D_HEAD + (q0 + r) * QKV_LD + 8 * hf;
#pragma unroll
    for (int kk = 0; kk < 4; ++kk) load_f8(qfb[kk], pQ, 16 * kk);
  }
  const float* pK = qkv + C_DIM + hh * D_HEAD + r * QKV_LD + 8 * hf;
  const float* pV = vnew + (size_t)hh * T_SEQ * D_HEAD + 8 * hf * D_HEAD + r;

  auto LOADK = [&]() {
#pragma unroll
    for (int kk = 0; kk < 4; ++kk) load_f8(kfb[kk], pK, 16 * kk);
  };
  LOADK();                                   // tile jt=0

  for (int jt = 0; jt <= qt; ++jt) {
    int j0 = jt << 4;
    v8f s = vz;
#pragma unroll
    for (int kk = 0; kk < 4; ++kk)
#pragma unroll
      for (int ss = 0; ss < 4; ++ss)
        s = wmma_f32(f8_frag(qfb[kk], ss), f8_frag(kfb[kk], ss), s);

    float sv[8];
#pragma unroll
    for (int i = 0; i < 8; ++i) {
      float v = s[i] * SCALE;
      if (jt == qt && (j0 + r) > (q0 + i + 8 * hf)) v = -3.4e38f;   // causal
      sv[i] = v;
    }
    float pv[8];
#pragma unroll
    for (int i = 0; i < 8; ++i) {
      float rmax = rowred_max(sv[i]);
      float mnew = fmaxf(mrow[i], rmax);
      float alpha = __expf(mrow[i] - mnew);
      pv[i] = __expf(sv[i] - mnew);
      float rs = rowred_sum(pv[i]);
      lrow[i] = lrow[i] * alpha + rs;
      mrow[i] = mnew;
#pragma unroll
      for (int t = 0; t < 4; ++t) acc[t][i] *= alpha;
    }
#pragma unroll
    for (int i = 0; i < 8; ++i) P[i + 8 * hf][r] = pv[i];
    asm volatile("s_wait_dscnt 0" ::: "memory");

    pK += 16 * QKV_LD;
    if (jt < qt) LOADK();                   // hide next tile behind P@V

#pragma unroll
    for (int ss = 0; ss < 4; ++ss) {
      v2f pa = *(const v2f*)&P[r][8 * hf + 2 * ss];   // A-frag of P
#pragma unroll
      for (int t = 0; t < 4; ++t) {
        v2f b;
        b.x = pV[ss * 2 * D_HEAD + t * 16];
        b.y = pV[ss * 2 * D_HEAD + t * 16 + D_HEAD];
        acc[t] = wmma_f32(pa, b, acc[t]);
      }
    }
    pV += 16 * D_HEAD;
    asm volatile("s_wait_dscnt 0" ::: "memory");
  }

  float* pY = y + (q0 + 8 * hf) * C_DIM + hh * D_HEAD + r;
#pragma unroll
  for (int t = 0; t < 4; ++t)
#pragma unroll
    for (int i = 0; i < 8; ++i)
      pY[i * C_DIM + t * 16] = acc[t][i] / lrow[i];
}

// ---------------------------------------------------------------------------
// launch
// ---------------------------------------------------------------------------
extern "C" void kernel_launch(void* const* d_in, const int* in_sizes, int n_in,
                              void* d_out, int out_size, void* d_ws, size_t ws_size,
                              hipStream_t stream) {
  (void)in_sizes; (void)n_in; (void)out_size; (void)ws_size;
  const float* x       = (const float*)d_in[0];
  const float* c_attn  = (const float*)d_in[1];
  const float* c_proj  = (const float*)d_in[2];
  const float* kstore  = (const float*)d_in[3];
  const float* vstore  = (const float*)d_in[4];
  float* out = (float*)d_out;

  float* ws   = (float*)d_ws;
  float* qkv  = ws;
  float* vnew = qkv  + (size_t)T_SEQ * QKV_LD;
  float* yb   = vnew + (size_t)H_NUM * T_SEQ * D_HEAD;
  float* ksq  = yb   + (size_t)T_SEQ * C_DIM;
  int*   mask = (int*)(ksq + (size_t)H_NUM * M_STORE);
  int*   idx  = mask + H_NUM * T_SEQ;

  ksq_kernel<<<(H_NUM * M_STORE + 255) / 256, 256, 0, stream>>>(
      kstore, ksq, H_NUM * M_STORE);

  {   // qkv = x @ Wqkv^T   (2048 x 3072 x 1024), 32x64 tiles
    int waves = (T_SEQ / 32) * (QKV_LD / 64);           // 3072
    gemm_xwT<<<waves / 8, 256, 0, stream>>>(x, c_attn, qkv,
        T_SEQ, QKV_LD, C_DIM, C_DIM, C_DIM, QKV_LD);
  }

  last_att_mask<<<H_NUM, 256, 0, stream>>>(qkv, mask);

  knn_topk<<<(H_NUM * (T_SEQ / 16)) / 8, 256, 0, stream>>>(qkv, kstore, ksq, idx);

  knn_vnew<<<(H_NUM * T_SEQ) / 8, 256, 0, stream>>>(qkv, kstore, vstore,
                                                    idx, mask, vnew);

  flash_attn<<<(H_NUM * (T_SEQ / 16)) / 8, 256, 0, stream>>>(qkv, vnew, yb);

  {   // out = y @ Wproj^T  (2048 x 1024 x 1024)
    int waves = (T_SEQ / 32) * (C_DIM / 64);            // 1024
    gemm_xwT<<<waves / 8, 256, 0, stream>>>(yb, c_proj, out,
        T_SEQ, C_DIM, C_DIM, C_DIM, C_DIM, C_DIM);
  }
}